// SdfParseLoss_5669356834128
// MI455X (gfx1250) — compile-verified
//
#include <hip/hip_runtime.h>
#include <hip/hip_bf16.h>

// Image geometry from the reference
#define W_IMG 384
#define H_IMG 512
#define HW_IMG (W_IMG * H_IMG)     // 196608
#define INF_VAL 9999.0f
#define NCHUNK 8                   // reduction chunks per batch
#define CHUNK_PX (HW_IMG / NCHUNK) // 24576
#define CHUNK_ITERS (CHUNK_PX / (8 * 64))  // 48 iterations, compile-time

typedef __attribute__((ext_vector_type(2))) float v2f;
typedef __attribute__((ext_vector_type(8))) float v8f;

// Monotone float<->uint encoding so unsigned atomicMin/Max implement float min/max.
__device__ __forceinline__ unsigned enc_f32(float f) {
    unsigned u = __float_as_uint(f);
    return (u & 0x80000000u) ? ~u : (u | 0x80000000u);
}
__device__ __forceinline__ float dec_f32(unsigned u) {
    unsigned b = (u & 0x80000000u) ? (u ^ 0x80000000u) : ~u;
    return __uint_as_float(b);
}

// ---------------------------------------------------------------------------
// Kernel 1: init scatter buffers (uint4 stores, 4 px / thread).
// min buffer starts at enc(+9999) so the reference's min(seg, 9999) clamp is
// implicit; max buffer at enc(-9999).
// ---------------------------------------------------------------------------
__global__ void init_bufs(uint4* __restrict__ minbuf4,
                          uint4* __restrict__ maxbuf4, int total4) {
    int i = blockIdx.x * blockDim.x + threadIdx.x;
    if (i < total4) {
        unsigned lo = enc_f32(INF_VAL);     // constant-folded
        unsigned hi = enc_f32(-INF_VAL);
        minbuf4[i] = make_uint4(lo, lo, lo, lo);
        maxbuf4[i] = make_uint4(hi, hi, hi, hi);
    }
}

// ---------------------------------------------------------------------------
// Kernel 2: scatter-min/max. Invalid points scatter +/-INF in the reference
// (neutral), so we skip them.
// ---------------------------------------------------------------------------
__global__ void scatter_minmax(const float* __restrict__ sdf,
                               const float* __restrict__ cm,
                               unsigned* __restrict__ minbuf,
                               unsigned* __restrict__ maxbuf, int N) {
    int n = blockIdx.x * blockDim.x + threadIdx.x;
    int b = blockIdx.y;
    if (n >= N) return;
    size_t i = (size_t)b * N + n;
    float x = cm[3 * i + 0];
    float y = cm[3 * i + 1];
    int ix = (int)x;   // trunc toward zero, matches .astype(int32)
    int iy = (int)y;
    if (ix >= 0 && ix < W_IMG && iy >= 0 && iy < H_IMG) {
        unsigned u = enc_f32(sdf[i]);
        size_t p = (size_t)b * HW_IMG + (size_t)(iy * W_IMG + ix);
        atomicMin(&minbuf[p], u);   // global_atomic_min_u32
        atomicMax(&maxbuf[p], u);   // global_atomic_max_u32
    }
}

// ---------------------------------------------------------------------------
// Kernel 3: partial loss reduction. Grid = (NCHUNK, B); one block = 8 wave32s,
// each wave eats 64 consecutive pixels per step (48 uniform steps).
// Masked dot product sum(t*u) accumulated with V_WMMA_F32_16X16X4_F32:
//   A (16x4)  row m, k = t[4m+k]   (lanes 0-15: K=0/1, lanes 16-31: K=2/3)
//   B (4x16)  col n, k = u[4n+k]   (symmetric lane split)
//   => D[m,m] += sum_k t[4m+k]*u[4m+k]; trace(D) = tile sum.
// Both |min'| and |max'-th| are blended with 0/1 float masks (no branches:
// keeps loads hoisted/coalesced and EXEC all-ones through every WMMA).
// ---------------------------------------------------------------------------
__global__ void __launch_bounds__(256) reduce_partial(
    const unsigned* __restrict__ minbuf, const unsigned* __restrict__ maxbuf,
    const int* __restrict__ gt, const float* __restrict__ valid,
    const float* __restrict__ thr_p,
    float* __restrict__ psum, int* __restrict__ pcnt) {
    const int chunk = blockIdx.x;     // 0..NCHUNK-1
    const int b = blockIdx.y;
    const int tid = threadIdx.x;
    const int wave = tid >> 5;
    const int lane = tid & 31;
    const float th = thr_p[0];

    const unsigned* mb = minbuf + (size_t)b * HW_IMG;
    const unsigned* xb = maxbuf + (size_t)b * HW_IMG;
    const int* g = gt + (size_t)b * HW_IMG;
    const float* vv = valid + (size_t)b * HW_IMG;

    // even per-lane pixel offset inside a 64-pixel tile -> 8B-aligned pairs
    const int q0 = ((lane & 15) << 2) | ((lane >> 4) << 1);
    const int base0 = chunk * CHUNK_PX + wave * 64 + q0;

    v8f c = {0.f, 0.f, 0.f, 0.f, 0.f, 0.f, 0.f, 0.f};
    int poscnt = 0;

#pragma unroll 2
    for (int it = 0; it < CHUNK_ITERS; ++it) {   // uniform trip count
        const int p = base0 + it * (8 * 64);
        __builtin_prefetch(&mb[p + 8 * 64], 0, 3);  // global_prefetch_b8 (near)
        __builtin_prefetch(&xb[p + 8 * 64], 0, 3);
        uint2  mnu = *(const uint2*)(mb + p);       // global_load_b64
        uint2  mxu = *(const uint2*)(xb + p);
        int2   gg  = *(const int2*)(g + p);
        float2 uu  = *(const float2*)(vv + p);

        float mn0 = dec_f32(mnu.x), mn1 = dec_f32(mnu.y);
        float mx0 = dec_f32(mxu.x), mx1 = dec_f32(mxu.y);
        // empty-bucket fixups (cndmask, no divergence)
        mn0 = (mn0 == INF_VAL) ? 0.f : mn0;
        mn1 = (mn1 == INF_VAL) ? 0.f : mn1;
        mx0 = (mx0 == -INF_VAL) ? th : mx0;
        mx1 = (mx1 == -INF_VAL) ? th : mx1;
        // 0/1 float masks: both loaded streams are used unconditionally,
        // so the backend cannot sink the loads into divergent blocks.
        float pos0 = (gg.x == 1) ? 1.f : 0.f;
        float pos1 = (gg.y == 1) ? 1.f : 0.f;
        float neg0 = (gg.x == 0) ? 1.f : 0.f;
        float neg1 = (gg.y == 0) ? 1.f : 0.f;
        float t0 = pos0 * fabsf(mn0) + neg0 * fabsf(mx0 - th);
        float t1 = pos1 * fabsf(mn1) + neg1 * fabsf(mx1 - th);
        poscnt += (gg.x == 1) + (gg.y == 1);

        v2f a;  a.x = t0;   a.y = t1;
        v2f bb; bb.x = uu.x; bb.y = uu.y;
        c = __builtin_amdgcn_wmma_f32_16x16x4_f32(
            /*neg_a=*/false, a, /*neg_b=*/false, bb,
            /*c_mod=*/(short)0, c, /*reuse_a=*/false, /*reuse_b=*/false);
    }

    // trace(D): C/D layout -> VGPR r, lane l<16 holds (M=r,N=l);
    // lane l>=16 holds (M=r+8,N=l-16). Diagonal at lane==r and lane==r+24.
    float diag = 0.f;
#pragma unroll
    for (int r = 0; r < 8; ++r) {
        bool on = (lane == r) || (lane == r + 24);
        diag += on ? c[r] : 0.f;
    }
#pragma unroll
    for (int off = 16; off > 0; off >>= 1) {
        diag += __shfl_xor(diag, off, 32);
        poscnt += __shfl_xor(poscnt, off, 32);
    }

    __shared__ float ssum[8];
    __shared__ int scnt[8];
    if (lane == 0) { ssum[wave] = diag; scnt[wave] = poscnt; }
    __syncthreads();
    if (tid == 0) {
        float s = 0.f; int cc = 0;
#pragma unroll
        for (int w = 0; w < 8; ++w) { s += ssum[w]; cc += scnt[w]; }
        psum[b * NCHUNK + chunk] = s;
        pcnt[b * NCHUNK + chunk] = cc;
    }
}

// ---------------------------------------------------------------------------
// Kernel 4: deterministic finalize (fixed-order partial combine).
// ---------------------------------------------------------------------------
__global__ void finalize_loss(const float* __restrict__ psum,
                              const int* __restrict__ pcnt,
                              float* __restrict__ out, int B) {
    int b = blockIdx.x * blockDim.x + threadIdx.x;
    if (b < B) {
        float s = 0.f; int cc = 0;
#pragma unroll
        for (int k = 0; k < NCHUNK; ++k) {
            s += psum[b * NCHUNK + k];
            cc += pcnt[b * NCHUNK + k];
        }
        out[b] = (cc > 0) ? (s * (1.0f / (float)HW_IMG)) : 0.f;
    }
}

// ---------------------------------------------------------------------------
extern "C" void kernel_launch(void* const* d_in, const int* in_sizes, int n_in,
                              void* d_out, int out_size, void* d_ws, size_t ws_size,
                              hipStream_t stream) {
    const float* sdf   = (const float*)d_in[0];   // (B,N)
    const float* cm    = (const float*)d_in[1];   // (B,N,3)
    const int*   gt    = (const int*)d_in[2];     // (B,H,W)
    const float* thr   = (const float*)d_in[3];   // scalar
    const float* valid = (const float*)d_in[5];   // (B,H,W)

    const int B = in_sizes[2] / HW_IMG;
    const int N = in_sizes[0] / B;

    unsigned* minbuf = (unsigned*)d_ws;                       // B*HW u32
    unsigned* maxbuf = minbuf + (size_t)B * HW_IMG;           // B*HW u32
    float*    psum   = (float*)(maxbuf + (size_t)B * HW_IMG); // B*NCHUNK f32
    int*      pcnt   = (int*)(psum + (size_t)B * NCHUNK);     // B*NCHUNK i32

    const int total4 = (B * HW_IMG) / 4;
    init_bufs<<<(total4 + 255) / 256, 256, 0, stream>>>(
        (uint4*)minbuf, (uint4*)maxbuf, total4);

    dim3 sg((N + 255) / 256, B);
    scatter_minmax<<<sg, 256, 0, stream>>>(sdf, cm, minbuf, maxbuf, N);

    dim3 rg(NCHUNK, B);
    reduce_partial<<<rg, 256, 0, stream>>>(minbuf, maxbuf, gt, valid, thr,
                                           psum, pcnt);

    finalize_loss<<<(B + 63) / 64, 64, 0, stream>>>(psum, pcnt,
                                                    (float*)d_out, B);
}